// Model_55301998903630
// MI455X (gfx1250) — compile-verified
//
#include <hip/hip_runtime.h>
#include <hip/hip_bf16.h>

typedef __attribute__((ext_vector_type(2))) float v2f;
typedef __attribute__((ext_vector_type(8))) float v8f;

#define Bsz  64
#define Tlen 512
#define Hdim 768
#define Ltag 9

// workspace layout (in floats)
#define EMIS_OFF    0
#define EMIS_ELEMS  (Bsz * Tlen * 16)          // padded [B*T][16]
#define LLH_OFF     (EMIS_OFF + EMIS_ELEMS)    // 64
#define MSUM_OFF    (LLH_OFF + 64)             // 64
#define BFRAG_OFF   (MSUM_OFF + 64)            // 192 k-steps * 64 lanes * 2 = 24576

// ---------------------------------------------------------------------------
// Pre-swizzle W (H x L, row-major) into per-lane WMMA B-fragment layout.
// B frag for k-step k4 (covers K = 4*k4 .. 4*k4+7):
//   frag[v] of lane l = W[4*k4 + 2*half + v][n],  half = l>>4, n = l&15
// stored at bfrag[k4*128 + l*2 + v]  (cols >= 9 zero-padded)
// ---------------------------------------------------------------------------
__global__ __launch_bounds__(256) void prep_bfrag(const float* __restrict__ W,
                                                  float* __restrict__ bfrag) {
    int idx = blockIdx.x * 256 + threadIdx.x;
    if (idx >= 192 * 128) return;
    int k4   = idx >> 7;
    int rem  = idx & 127;
    int l    = rem >> 1;
    int v    = rem & 1;
    int half = l >> 4;
    int n    = l & 15;
    int r    = 4 * k4 + 2 * half + v;        // 0..771
    bfrag[idx] = (n < Ltag) ? W[r * Ltag + n] : 0.0f;
}

// ---------------------------------------------------------------------------
// Emissions GEMM: emis[row][n] = hidden[row][:] . W[:][n] + b[n]
// 256 threads = 8 waves; each wave owns one 16x16 WMMA tile; block tile = 128
// rows. A is staged with GLOBAL_LOAD_ASYNC_TO_LDS_B128 (CDNA5 async DMA path,
// tracked by ASYNCcnt) into a double-buffered LDS tile (row stride padded
// 32->36 floats for conflict-free ds_load_b64 fragment reads). No VGPR
// staging -> no spills. 192 V_WMMA_F32_16X16X4_F32 per wave.
// ---------------------------------------------------------------------------
#define LDS_STRIDE 36
#define BUF_BYTES  (128 * LDS_STRIDE * 4)     // 18432

__global__ __launch_bounds__(256) void emis_gemm(const float* __restrict__ hid,
                                                 const float* __restrict__ bias,
                                                 const float* __restrict__ bfrag,
                                                 float* __restrict__ emis) {
    __shared__ float As[2][128 * LDS_STRIDE];

    const int tid   = threadIdx.x;
    const int w     = tid >> 5;          // wave id 0..7
    const int lane  = tid & 31;
    const int half  = lane >> 4;
    const int n     = lane & 15;
    const int mBase = blockIdx.x * 128;

    v8f acc = {};

    // Per-thread async-copy slots: 4 x b128 per stage (128 rows x 32 floats).
    // voff = per-lane global byte offset (GVS mode), lds0 = LDS byte offset
    // in buffer 0; buffer 1 adds BUF_BYTES.
    const unsigned ldsBase = (unsigned)(uintptr_t)&As[0][0];
    unsigned voff[4], lds0[4];
#pragma unroll
    for (int i = 0; i < 4; ++i) {
        int idx = tid + i * 256;         // 0..1023 float4 slots
        int row = idx >> 3;              // 128 rows, 8 float4 per row
        int c4  = idx & 7;
        voff[i] = (unsigned)((row * Hdim + (c4 << 2)) << 2);
        lds0[i] = ldsBase + (unsigned)(((row * LDS_STRIDE + (c4 << 2)) << 2));
    }

    const float* gbase = hid + (size_t)mBase * Hdim;

    // prologue: async-fill stage 0 into buffer 0
    {
        const float* gb = gbase;         // k0 = 0
#pragma unroll
        for (int i = 0; i < 4; ++i) {
            asm volatile("global_load_async_to_lds_b128 %0, %1, %2"
                         :: "v"(lds0[i]), "v"(voff[i]), "s"(gb)
                         : "memory");
        }
    }

    for (int s = 0; s < 24; ++s) {       // 24 stages of Kc=32
        const int buf = s & 1;

        // wait for this stage's async DMA, then make it visible block-wide
        asm volatile("s_wait_asynccnt 0x0" ::: "memory");
        __syncthreads();

        if (s < 23) {                    // kick off next stage into other buffer
            const float* gb = gbase + ((s + 1) << 5);
            const unsigned bb = (buf ^ 1) ? (unsigned)BUF_BYTES : 0u;
#pragma unroll
            for (int i = 0; i < 4; ++i) {
                asm volatile("global_load_async_to_lds_b128 %0, %1, %2"
                             :: "v"(lds0[i] + bb), "v"(voff[i]), "s"(gb)
                             : "memory");
            }
        }

        // A fragment base: row = w*16 + (lane&15), col = kl + 2*half
        const float* Ab = &As[buf][((w << 4) + (lane & 15)) * LDS_STRIDE + (half << 1)];
#pragma unroll
        for (int kl = 0; kl < 32; kl += 4) {
            v2f a  = *(const v2f*)(Ab + kl);
            v2f bb = *(const v2f*)&bfrag[(((s << 3) + (kl >> 2)) << 6) + (lane << 1)];
            acc = __builtin_amdgcn_wmma_f32_16x16x4_f32(
                /*neg_a=*/false, a, /*neg_b=*/false, bb,
                /*c_mod=*/(short)0, acc, /*reuse_a=*/false, /*reuse_b=*/false);
        }
        __syncthreads();                 // all waves done before buffer reuse
    }

    // C/D layout: VGPR v, lane -> row = v + 8*half, col = lane&15
    float bn = (n < Ltag) ? bias[n] : 0.0f;
    int row0 = mBase + (w << 4) + (half << 3);
#pragma unroll
    for (int v = 0; v < 8; ++v) {
        emis[(size_t)(row0 + v) * 16 + n] = acc[v] + bn;
    }
}

// ---------------------------------------------------------------------------
// CRF per-batch: forward algorithm (sequential over T) + gold-path numerator
// (parallel over T). One wave per batch. Lane j owns next-tag column j; alpha
// replicated in every lane so the L=9 logsumexp needs no cross-lane traffic,
// only 9 broadcasts per step.
// ---------------------------------------------------------------------------
__global__ __launch_bounds__(32) void crf_scan(const float* __restrict__ emis,
                                               const float* __restrict__ startT,
                                               const float* __restrict__ endT,
                                               const float* __restrict__ trans,
                                               const int*   __restrict__ tags,
                                               const int*   __restrict__ mask,
                                               float* __restrict__ llh,
                                               float* __restrict__ msum) {
    const int b    = blockIdx.x;
    const int lane = threadIdx.x;

    __shared__ float tS[81];
    for (int i = lane; i < 81; i += 32) tS[i] = trans[i];
    __syncthreads();

    const int jc = (lane < Ltag) ? lane : (Ltag - 1);
    float trCol[Ltag];
#pragma unroll
    for (int i = 0; i < Ltag; ++i) trCol[i] = tS[i * Ltag + jc];

    const float* emB   = emis + (size_t)b * Tlen * 16;
    const int*   maskB = mask + b * Tlen;
    const int*   tagB  = tags + b * Tlen;

    float alpha[Ltag];
#pragma unroll
    for (int i = 0; i < Ltag; ++i) alpha[i] = startT[i] + emB[i];

    // ---- forward recursion ----
    for (int t = 1; t < Tlen; ++t) {
        int   m   = maskB[t];
        float emj = emB[t * 16 + jc];
        float mx  = -1e30f;
#pragma unroll
        for (int i = 0; i < Ltag; ++i) mx = fmaxf(mx, alpha[i] + trCol[i]);
        float sm = 0.0f;
#pragma unroll
        for (int i = 0; i < Ltag; ++i) sm += __expf(alpha[i] + trCol[i] - mx);
        float aj = mx + __logf(sm) + emj;
        if (m == 0) aj = alpha[jc];
#pragma unroll
        for (int i = 0; i < Ltag; ++i) alpha[i] = __shfl(aj, i, 32);
    }

    // ---- mask total ----
    int ms = 0;
    for (int t = lane; t < Tlen; t += 32) ms += maskB[t];
#pragma unroll
    for (int o = 16; o > 0; o >>= 1) ms += __shfl_xor(ms, o, 32);

    // ---- numerator (parallel over t) ----
    float num = 0.0f;
    for (int t = lane + 1; t < Tlen; t += 32) {
        int m = maskB[t];
        if (m) {
            int tp = tagB[t - 1];
            int tc = tagB[t];
            num += tS[tp * Ltag + tc] + emB[t * 16 + tc];
        }
    }
#pragma unroll
    for (int o = 16; o > 0; o >>= 1) num += __shfl_xor(num, o, 32);

    // ---- denominator: logsumexp(alpha + endT)  (alpha replicated, no comms) ----
    float mx = -1e30f;
#pragma unroll
    for (int i = 0; i < Ltag; ++i) mx = fmaxf(mx, alpha[i] + endT[i]);
    float sm = 0.0f;
#pragma unroll
    for (int i = 0; i < Ltag; ++i) sm += __expf(alpha[i] + endT[i] - mx);
    float den = mx + __logf(sm);

    if (lane == 0) {
        int   tag0 = tagB[0];
        float tot  = num + startT[tag0] + emB[tag0];    // score at t=0
        int   last = tagB[ms - 1];                      // seq_end = sum(mask)-1
        tot += endT[last];
        llh[b]  = tot - den;
        msum[b] = (float)ms;
    }
}

// ---------------------------------------------------------------------------
// Final reduction: loss = -(sum(llh) / sum(mask))
// ---------------------------------------------------------------------------
__global__ __launch_bounds__(32) void finalize(const float* __restrict__ llh,
                                               const float* __restrict__ msum,
                                               float* __restrict__ out) {
    int lane = threadIdx.x;
    float v  = llh[lane]  + llh[lane + 32];
    float ms = msum[lane] + msum[lane + 32];
#pragma unroll
    for (int o = 16; o > 0; o >>= 1) {
        v  += __shfl_xor(v,  o, 32);
        ms += __shfl_xor(ms, o, 32);
    }
    if (lane == 0) out[0] = -(v / ms);
}

extern "C" void kernel_launch(void* const* d_in, const int* in_sizes, int n_in,
                              void* d_out, int out_size, void* d_ws, size_t ws_size,
                              hipStream_t stream) {
    const float* hidden = (const float*)d_in[0];   // [B,T,H] f32
    const float* tw     = (const float*)d_in[1];   // [H,L]   f32
    const float* tb     = (const float*)d_in[2];   // [L]     f32
    const float* st     = (const float*)d_in[3];   // [L]     f32
    const float* et     = (const float*)d_in[4];   // [L]     f32
    const float* tr     = (const float*)d_in[5];   // [L,L]   f32
    const int*   tags   = (const int*)d_in[6];     // [B,T]
    const int*   mask   = (const int*)d_in[7];     // [B,T]

    float* ws    = (float*)d_ws;
    float* emis  = ws + EMIS_OFF;
    float* llh   = ws + LLH_OFF;
    float* ms    = ws + MSUM_OFF;
    float* bfrag = ws + BFRAG_OFF;

    prep_bfrag<<<(192 * 128 + 255) / 256, 256, 0, stream>>>(tw, bfrag);
    emis_gemm<<<(Bsz * Tlen) / 128, 256, 0, stream>>>(hidden, tb, bfrag, emis);
    crf_scan<<<Bsz, 32, 0, stream>>>(emis, st, et, tr, tags, mask, llh, ms);
    finalize<<<1, 32, 0, stream>>>(llh, ms, (float*)d_out);
}